// MoELoRA_19679540150609
// MI455X (gfx1250) — compile-verified
//
#include <hip/hip_runtime.h>
#include <math.h>

// ---- problem constants (match reference) ----
#define B_   4
#define S_   2048
#define D_   4096
#define E_   4
#define R_   64
#define OUT_ 4096
#define HID_ 128
#define SCALING_ 2.0f          // ALPHA / R = 128/64
#define BALANCE_COEFF_ 0.01f

typedef float v2f __attribute__((ext_vector_type(2)));
typedef float v8f __attribute__((ext_vector_type(8)));

// ---------------------------------------------------------------------------
// Kernel 1: mean-pool x over S.  pooled[b][d] = mean_s x[b][s][d]
// grid (D/256, B), block 256 — loads are coalesced across d each iteration.
// ---------------------------------------------------------------------------
__global__ void moelora_pool(const float* __restrict__ x, float* __restrict__ pooled) {
    const int b = blockIdx.y;
    const int d = blockIdx.x * blockDim.x + threadIdx.x;
    const float* xp = x + (size_t)b * S_ * D_ + d;
    float s = 0.0f;
    #pragma unroll 8
    for (int i = 0; i < S_; ++i) s += xp[(size_t)i * D_];
    pooled[b * D_ + d] = s * (1.0f / (float)S_);
}

// ---------------------------------------------------------------------------
// Kernel 2: router MLP on [pooled | reliability | regime] -> gelu -> logits
// -> softmax/argmax -> top_expert[b], balance loss.  Single block, 128 thr.
// ---------------------------------------------------------------------------
__global__ void moelora_router(const float* __restrict__ pooled,
                               const float* __restrict__ rel,
                               const float* __restrict__ reg,
                               const float* __restrict__ w1, const float* __restrict__ b1,
                               const float* __restrict__ w2, const float* __restrict__ b2,
                               int* __restrict__ topE, float* __restrict__ loss_out) {
    __shared__ float h[B_][HID_];
    __shared__ float logits[B_][E_];
    __shared__ float probs[B_][E_];
    const int t = threadIdx.x;              // 0..127

    for (int b = 0; b < B_; ++b) {
        float acc = b1[t];
        const float* pb = pooled + b * D_;
        for (int i = 0; i < D_; ++i) acc += pb[i] * w1[i * HID_ + t];
        for (int i = 0; i < 4; ++i)  acc += rel[b * 4 + i] * w1[(D_ + i) * HID_ + t];
        for (int i = 0; i < 3; ++i)  acc += reg[b * 3 + i] * w1[(D_ + 4 + i) * HID_ + t];
        // jax.nn.gelu default: tanh approximation
        float xg = acc;
        float g = 0.5f * xg * (1.0f + tanhf(0.7978845608028654f * (xg + 0.044715f * xg * xg * xg)));
        h[b][t] = g;
    }
    __syncthreads();

    if (t < B_ * E_) {
        const int b = t >> 2, e = t & 3;
        float acc = b2[e];
        for (int j = 0; j < HID_; ++j) acc += h[b][j] * w2[j * E_ + e];
        logits[b][e] = acc;
    }
    __syncthreads();

    if (t < B_) {
        const int b = t;
        float m = logits[b][0]; int arg = 0;
        for (int e = 1; e < E_; ++e) if (logits[b][e] > m) { m = logits[b][e]; arg = e; }
        float p[E_], s = 0.0f;
        for (int e = 0; e < E_; ++e) { p[e] = expf(logits[b][e] - m); s += p[e]; }
        const float inv = 1.0f / s;
        for (int e = 0; e < E_; ++e) probs[b][e] = p[e] * inv;
        topE[b] = arg;
    }
    __syncthreads();

    if (t == 0) {
        float loss = 0.0f;
        for (int e = 0; e < E_; ++e) {
            float a = 0.0f;
            for (int b = 0; b < B_; ++b) a += probs[b][e];
            a *= (1.0f / (float)B_);
            loss += a * a;
        }
        loss_out[0] = BALANCE_COEFF_ * (float)E_ * loss;
    }
}

// ---------------------------------------------------------------------------
// GEMM1: xr[b][s][r] = sum_d x[b][s][d] * lora_A[e][r][d]   (M=S,N=R,K=D)
// One wave computes a 16(M) x 64(N) tile; K loop in steps of 4 via
// V_WMMA_F32_16X16X4_F32.  A-frag and B-frag use the identical per-lane
// addressing: element(vgpr j) = Mat[lane&15][k0 + 2*(lane>>4) + j].
// ---------------------------------------------------------------------------
__global__ void moelora_gemm1(const float* __restrict__ x,
                              const float* __restrict__ loraA,
                              const int* __restrict__ topE,
                              float* __restrict__ xr) {
    const int lane = threadIdx.x & 31;
    const int wave = threadIdx.x >> 5;
    const int b    = blockIdx.y;
    const int e    = topE[b];
    const int m0   = (blockIdx.x * 4 + wave) * 16;     // 4 waves / block

    const float* X = x + ((size_t)b * S_ + m0) * D_;
    const float* A = loraA + (size_t)e * R_ * D_;

    const int row  = lane & 15;           // M row (A-frag) / N col (B-frag)
    const int koff = (lane >> 4) * 2;     // K offset within 4-wide step

    v8f acc0 = {}, acc1 = {}, acc2 = {}, acc3 = {};

    const float* xrow = X + (size_t)row * D_ + koff;
    const float* a0 = A + (size_t)(0 * 16 + row) * D_ + koff;
    const float* a1 = A + (size_t)(1 * 16 + row) * D_ + koff;
    const float* a2 = A + (size_t)(2 * 16 + row) * D_ + koff;
    const float* a3 = A + (size_t)(3 * 16 + row) * D_ + koff;

    #pragma unroll 4
    for (int k0 = 0; k0 < D_; k0 += 4) {
        v2f av = *(const v2f*)(xrow + k0);
        v2f b0 = *(const v2f*)(a0 + k0);
        v2f b1 = *(const v2f*)(a1 + k0);
        v2f b2 = *(const v2f*)(a2 + k0);
        v2f b3 = *(const v2f*)(a3 + k0);
        acc0 = __builtin_amdgcn_wmma_f32_16x16x4_f32(false, av, false, b0, (short)0, acc0, false, false);
        acc1 = __builtin_amdgcn_wmma_f32_16x16x4_f32(false, av, false, b1, (short)0, acc1, false, false);
        acc2 = __builtin_amdgcn_wmma_f32_16x16x4_f32(false, av, false, b2, (short)0, acc2, false, false);
        acc3 = __builtin_amdgcn_wmma_f32_16x16x4_f32(false, av, false, b3, (short)0, acc3, false, false);
    }

    // C/D layout: vgpr j, lane L -> (M = 8*(L>>4) + j, N = L&15)
    float* out = xr + ((size_t)b * S_ + m0) * R_;
    const int rbase = (lane >> 4) * 8;
    const int col   = lane & 15;
    #pragma unroll
    for (int j = 0; j < 8; ++j) {
        const size_t r = (size_t)(rbase + j) * R_;
        out[r + 0 * 16 + col] = acc0[j];
        out[r + 1 * 16 + col] = acc1[j];
        out[r + 2 * 16 + col] = acc2[j];
        out[r + 3 * 16 + col] = acc3[j];
    }
}

// ---------------------------------------------------------------------------
// GEMM2: out[b][s][o] = 2 * sum_r xr[b][s][r] * lora_B[e][o][r]  (M=S,N=OUT,K=R)
// One wave computes a 16 x 64 tile; 16 WMMA steps over K=64.
// ---------------------------------------------------------------------------
__global__ void moelora_gemm2(const float* __restrict__ xr,
                              const float* __restrict__ loraB,
                              const int* __restrict__ topE,
                              float* __restrict__ out) {
    const int lane = threadIdx.x & 31;
    const int wave = threadIdx.x >> 5;
    const int b    = blockIdx.z;
    const int e    = topE[b];
    const int m0   = blockIdx.y * 16;
    const int n0   = blockIdx.x * 512 + wave * 64;     // 8 waves / block

    const float* XR = xr + ((size_t)b * S_ + m0) * R_;
    const float* BW = loraB + (size_t)e * OUT_ * R_;

    const int row  = lane & 15;
    const int koff = (lane >> 4) * 2;

    v8f acc0 = {}, acc1 = {}, acc2 = {}, acc3 = {};

    const float* xrow = XR + (size_t)row * R_ + koff;
    const float* b0p = BW + (size_t)(n0 + 0 * 16 + row) * R_ + koff;
    const float* b1p = BW + (size_t)(n0 + 1 * 16 + row) * R_ + koff;
    const float* b2p = BW + (size_t)(n0 + 2 * 16 + row) * R_ + koff;
    const float* b3p = BW + (size_t)(n0 + 3 * 16 + row) * R_ + koff;

    #pragma unroll
    for (int k0 = 0; k0 < R_; k0 += 4) {
        v2f av = *(const v2f*)(xrow + k0);
        v2f b0 = *(const v2f*)(b0p + k0);
        v2f b1 = *(const v2f*)(b1p + k0);
        v2f b2 = *(const v2f*)(b2p + k0);
        v2f b3 = *(const v2f*)(b3p + k0);
        acc0 = __builtin_amdgcn_wmma_f32_16x16x4_f32(false, av, false, b0, (short)0, acc0, false, false);
        acc1 = __builtin_amdgcn_wmma_f32_16x16x4_f32(false, av, false, b1, (short)0, acc1, false, false);
        acc2 = __builtin_amdgcn_wmma_f32_16x16x4_f32(false, av, false, b2, (short)0, acc2, false, false);
        acc3 = __builtin_amdgcn_wmma_f32_16x16x4_f32(false, av, false, b3, (short)0, acc3, false, false);
    }

    float* op = out + ((size_t)b * S_ + m0) * (size_t)OUT_ + n0;
    const int rbase = (lane >> 4) * 8;
    const int col   = lane & 15;
    #pragma unroll
    for (int j = 0; j < 8; ++j) {
        const size_t r = (size_t)(rbase + j) * OUT_;
        op[r + 0 * 16 + col] = SCALING_ * acc0[j];
        op[r + 1 * 16 + col] = SCALING_ * acc1[j];
        op[r + 2 * 16 + col] = SCALING_ * acc2[j];
        op[r + 3 * 16 + col] = SCALING_ * acc3[j];
    }
}

// ---------------------------------------------------------------------------
extern "C" void kernel_launch(void* const* d_in, const int* in_sizes, int n_in,
                              void* d_out, int out_size, void* d_ws, size_t ws_size,
                              hipStream_t stream) {
    const float* x     = (const float*)d_in[0];
    const float* rel   = (const float*)d_in[1];
    const float* reg   = (const float*)d_in[2];
    const float* loraA = (const float*)d_in[3];
    const float* loraB = (const float*)d_in[4];
    const float* w1    = (const float*)d_in[5];
    const float* b1    = (const float*)d_in[6];
    const float* w2    = (const float*)d_in[7];
    const float* b2    = (const float*)d_in[8];

    float* out  = (float*)d_out;
    float* loss = out + (size_t)B_ * S_ * OUT_;   // scalar appended after out tensor

    // workspace layout
    float* ws     = (float*)d_ws;
    float* pooled = ws;                                  // B*D floats
    float* xr     = ws + (size_t)B_ * D_;                // B*S*R floats
    int*   topE   = (int*)(ws + (size_t)B_ * D_ + (size_t)B_ * S_ * R_);

    // 1) mean pool over sequence
    moelora_pool<<<dim3(D_ / 256, B_), 256, 0, stream>>>(x, pooled);

    // 2) router MLP -> top_expert + balance loss
    moelora_router<<<1, HID_, 0, stream>>>(pooled, rel, reg, w1, b1, w2, b2, topE, loss);

    // 3) xr = X @ A_e^T   (fp32 WMMA, M=2048 N=64 K=4096 per batch)
    moelora_gemm1<<<dim3(S_ / 16 / 4, B_), 128, 0, stream>>>(x, loraA, topE, xr);

    // 4) out = 2 * xr @ B_e^T  (fp32 WMMA, M=2048 N=4096 K=64 per batch)
    moelora_gemm2<<<dim3(OUT_ / 512, S_ / 16, B_), 256, 0, stream>>>(xr, loraB, topE, out);
}